// AdaptiveKNN_35467839931084
// MI455X (gfx1250) — compile-verified
//
#include <hip/hip_runtime.h>
#include <hip/hip_bf16.h>
#include <math.h>

// ---------------------------------------------------------------------------
// AdaptiveKNN for MI455X (gfx1250, wave32).
//   B=2, N=8192, K_BASE=32, K in [8,128], RADIUS=0.05, EPS=1e-8.
// Distance core G = P.P^T is a rank-3 GEMM -> V_WMMA_F32_16X16X4_F32 (fp32,
// so the 2.5e-3 radius^2 threshold is not destroyed by fp16 rounding).
// Top-k uses register-resident candidate lists + wave32 shfl butterflies:
// one barrier per extraction, ~130B LDS.
// ---------------------------------------------------------------------------

#define BB 2
#define NN 8192
#define KMAX 128
#define RADIUS_F 0.05f
#define EPS_F 1e-8f

typedef __attribute__((ext_vector_type(2))) float v2f;
typedef __attribute__((ext_vector_type(8))) float v8f;

// ---------------------------------------------------------------------------
// Kernel 1: local density via WMMA gram tiles.
// Block = 256 threads = 8 waves; wave w owns 16 query rows.
// A (16x4, loop invariant): row m -> (x, y, t, 1)
//   layout: lane L (m = L&15): VGPR0 = K0|K2, VGPR1 = K1|K3 (halves by L>=16)
// B (4x16, per j tile): col n -> (-2x, -2y, -2t, |p|^2)
// C = |p_j|^2 - 2*g(m,n);  d2 = C + |p_i|^2
// ---------------------------------------------------------------------------
__global__ __launch_bounds__(256) void adknn_density_kernel(
    const float* __restrict__ coords,   // [B,N,2]
    const float* __restrict__ times,    // [B,N]
    float* __restrict__ density)        // [B,N]
{
  const int b    = blockIdx.x / (NN / 128);
  const int tile = blockIdx.x % (NN / 128);
  const int wave = threadIdx.x >> 5;
  const int lane = threadIdx.x & 31;
  const int half = lane >> 4;          // 0: low K/M group, 1: high
  const int l    = lane & 15;
  const int i0   = tile * 128 + wave * 16;

  const float* cb = coords + (size_t)b * NN * 2;
  const float* tb = times  + (size_t)b * NN;

  // Loop-invariant A fragment (rows i0..i0+15).
  const int   m  = i0 + l;
  const float xm = cb[2 * m + 0];
  const float ym = cb[2 * m + 1];
  const float tm = tb[m];
  v2f afrag;
  afrag.x = half ? tm   : xm;   // K=2 : K=0
  afrag.y = half ? 1.0f : ym;   // K=3 : K=1

  // |p_i|^2 for my 8 C rows (row = i0 + half*8 + v).
  float si[8];
#pragma unroll
  for (int v = 0; v < 8; ++v) {
    const int r = i0 + half * 8 + v;
    const float x = cb[2 * r], y = cb[2 * r + 1], t = tb[r];
    si[v] = x * x + y * y + t * t;
  }

  float cnt[8];
#pragma unroll
  for (int v = 0; v < 8; ++v) cnt[v] = 0.0f;

  const float thr = RADIUS_F * RADIUS_F - EPS_F;  // d2 < thr  <=>  dist < R

  for (int j0 = 0; j0 < NN; j0 += 16) {
    const int jn = j0 + l;
    __builtin_prefetch(cb + 2 * (jn + 256), 0, 1);  // global_prefetch_b8
    const float xj = cb[2 * jn + 0];
    const float yj = cb[2 * jn + 1];
    const float tj = tb[jn];
    const float sj = xj * xj + yj * yj + tj * tj;
    v2f bfrag;
    bfrag.x = half ? (-2.0f * tj) : (-2.0f * xj);  // K=2 : K=0
    bfrag.y = half ? sj           : (-2.0f * yj);  // K=3 : K=1

    v8f c = {};
    c = __builtin_amdgcn_wmma_f32_16x16x4_f32(
        /*neg_a=*/false, afrag, /*neg_b=*/false, bfrag,
        /*c_mod=*/(short)0, c, /*reuse_a=*/false, /*reuse_b=*/false);

#pragma unroll
    for (int v = 0; v < 8; ++v) {
      float d2 = fmaxf(c[v] + si[v], 0.0f);
      cnt[v] += (d2 < thr) ? 1.0f : 0.0f;
    }
  }

  // Sum over the 16 columns: xor masks 1,2,4,8 stay within each 16-lane half.
#pragma unroll
  for (int v = 0; v < 8; ++v) {
    float s = cnt[v];
    s += __shfl_xor(s, 1, 32);
    s += __shfl_xor(s, 2, 32);
    s += __shfl_xor(s, 4, 32);
    s += __shfl_xor(s, 8, 32);
    cnt[v] = s;
  }

  if (l == 0) {
    const float inv = 1.0f / ((float)NN * RADIUS_F * RADIUS_F * RADIUS_F + EPS_F);
#pragma unroll
    for (int v = 0; v < 8; ++v)
      density[(size_t)b * NN + i0 + half * 8 + v] = cnt[v] * inv;
  }
}

// ---------------------------------------------------------------------------
// Kernel 2: per-batch mean density -> adaptive k per point.
// ---------------------------------------------------------------------------
__global__ __launch_bounds__(256) void adknn_kval_kernel(
    const float* __restrict__ density,       // [B,N]
    const float* __restrict__ gamma_param,   // [1]
    int* __restrict__ kvals)                 // [B,N]
{
  __shared__ float red[256];
  const int b = blockIdx.x;
  const int tid = threadIdx.x;
  const float* db = density + (size_t)b * NN;

  float s = 0.0f;
  for (int j = tid; j < NN; j += 256) s += db[j];
  red[tid] = s;
  __syncthreads();
  for (int k = 128; k > 0; k >>= 1) {
    if (tid < k) red[tid] += red[tid + k];
    __syncthreads();
  }
  const float mean  = red[0] / (float)NN;
  const float gamma = 1.0f / (1.0f + expf(-gamma_param[0]));

  for (int j = tid; j < NN; j += 256) {
    const float ratio = powf(mean / (db[j] + EPS_F), gamma);
    float kv = 32.0f * ratio;                 // K_BASE
    kv = fminf(fmaxf(kv, 8.0f), 128.0f);      // clip to [K_MIN, K_MAX]
    kvals[(size_t)b * NN + j] = (int)kv;      // trunc like astype(int32)
  }
}

// ---------------------------------------------------------------------------
// Kernel 3: top-128 smallest distances per query point.
// One 256-thread block per (b,i). Each thread owns the 32 candidates
// j = tid + 256*s, held in REGISTERS (fully unrolled) with an "extracted"
// bitmask. Per extraction: wave32 shfl_xor butterfly (no barrier) -> 8
// per-wave minima in a double-buffered 16-entry LDS array -> every thread
// folds the 8 entries redundantly (winner known block-wide, no broadcast).
// Exactly one __syncthreads per extraction; sdist LDS eliminated entirely.
// Stable lowest-index tie-breaking matches lax.top_k.
// ---------------------------------------------------------------------------
__global__ __launch_bounds__(256) void adknn_topk_kernel(
    const float* __restrict__ coords,   // [B,N,2]
    const float* __restrict__ times,    // [B,N]
    const int*   __restrict__ kvals,    // [B,N]
    int*   __restrict__ out_idx,        // [B,N,KMAX]
    float* __restrict__ out_mask,       // [B,N,KMAX] (1.0 / 0.0)
    float* __restrict__ out_dist)       // [B,N,KMAX]
{
  __shared__ float wmin[2][8];
  __shared__ int   widx[2][8];

  const int tid  = threadIdx.x;
  const int lane = tid & 31;
  const int wave = tid >> 5;
  const int q    = blockIdx.x;        // 0 .. B*N-1
  const int b    = q / NN;
  const int i    = q % NN;
  const float* cb = coords + (size_t)b * NN * 2;
  const float* tb = times  + (size_t)b * NN;

  const float qx = cb[2 * i + 0];
  const float qy = cb[2 * i + 1];
  const float qt = tb[i];
  const float INF = __builtin_inff();

  // Register-resident candidate distances for j = tid + 256*s.
  float d[32];
#pragma unroll
  for (int s = 0; s < 32; ++s) {
    const int j = tid + (s << 8);
    const float dx = cb[2 * j + 0] - qx;
    const float dy = cb[2 * j + 1] - qy;
    const float dt = tb[j] - qt;
    const float d2 = dx * dx + dy * dy + dt * dt;
    d[s] = (j == i) ? INF : sqrtf(d2 + EPS_F);
  }
  unsigned int taken = 0u;

  // Cached per-thread local minimum (ascending j => lowest index on ties).
  float lmin = INF;
  int   lidx = 0x7fffffff;
#pragma unroll
  for (int s = 0; s < 32; ++s) {
    const float v = d[s];
    if (v < lmin) { lmin = v; lidx = tid + (s << 8); }
  }

  const int myk = kvals[q];

  for (int t = 0; t < KMAX; ++t) {
    const int p = t & 1;  // double buffer: removes the WAR barrier

    // Wave32 butterfly min-reduce of (value, index).
    float bv = lmin;
    int   bi = lidx;
#pragma unroll
    for (int msk = 16; msk >= 1; msk >>= 1) {
      const float ov = __shfl_xor(bv, msk, 32);
      const int   oj = __shfl_xor(bi, msk, 32);
      if (ov < bv || (ov == bv && oj < bi)) { bv = ov; bi = oj; }
    }
    if (lane == 0) { wmin[p][wave] = bv; widx[p][wave] = bi; }
    __syncthreads();   // the only barrier per extraction

    // Every thread folds the 8 wave results -> winner known block-wide.
    float gv = wmin[p][0];
    int   gj = widx[p][0];
#pragma unroll
    for (int w = 1; w < 8; ++w) {
      const float ov = wmin[p][w];
      const int   oj = widx[p][w];
      if (ov < gv || (ov == gv && oj < gj)) { gv = ov; gj = oj; }
    }

    if (tid == 0) {
      const size_t o = (size_t)q * KMAX + t;
      out_idx[o]  = gj;
      out_dist[o] = gv;
      out_mask[o] = (t < myk) ? 1.0f : 0.0f;
    }

    // Owner invalidates its slot and rescans its 32 registers.
    if ((gj & 255) == tid) {
      taken |= 1u << (gj >> 8);
      lmin = INF;
      lidx = 0x7fffffff;
#pragma unroll
      for (int s = 0; s < 32; ++s) {
        const float v = d[s];
        const bool live = ((taken >> s) & 1u) == 0u;
        if (live && v < lmin) { lmin = v; lidx = tid + (s << 8); }
      }
    }
    // No barrier needed here: the 16-entry LDS buffers alternate per round,
    // and round t's reads complete before round t+2's writes because of the
    // barrier inside round t+1.
  }
}

// ---------------------------------------------------------------------------
// Launcher. Inputs: coords[B,N,2] f32, times[B,N] f32, features[B,N,64] f32
// (unused by reference), gamma_param[1] f32.
// Outputs (concatenated): indices int32 [B,N,128], mask [B,N,128],
// distances f32 [B,N,128].
// Workspace: density f32[B*N] then kvals int32[B*N] (128 KB total).
// ---------------------------------------------------------------------------
extern "C" void kernel_launch(void* const* d_in, const int* in_sizes, int n_in,
                              void* d_out, int out_size, void* d_ws, size_t ws_size,
                              hipStream_t stream) {
  const float* coords      = (const float*)d_in[0];
  const float* times       = (const float*)d_in[1];
  const float* gamma_param = (const float*)d_in[3];
  (void)in_sizes; (void)n_in; (void)out_size; (void)ws_size;

  float* density = (float*)d_ws;
  int*   kvals   = (int*)((char*)d_ws + sizeof(float) * (size_t)BB * NN);

  const size_t sect = (size_t)BB * NN * KMAX;
  int*   out_idx  = (int*)d_out;
  float* out_mask = (float*)d_out + sect;
  float* out_dist = (float*)d_out + 2 * sect;

  adknn_density_kernel<<<BB * (NN / 128), 256, 0, stream>>>(coords, times, density);
  adknn_kval_kernel<<<BB, 256, 0, stream>>>(density, gamma_param, kvals);
  adknn_topk_kernel<<<BB * NN, 256, 0, stream>>>(coords, times, kvals,
                                                 out_idx, out_mask, out_dist);
}